// HungarianMatcher_41240275976594
// MI455X (gfx1250) — compile-verified
//
#include <hip/hip_runtime.h>
#include <math.h>

// Problem constants (from reference)
#define BB 32
#define KK 8
#define AA 8
#define PP 16
#define CC 3
#define HWSZ 25600            // 160*160
#define OO 76800              // 3*160*160
#define NTILES 1600           // HWSZ / 16

typedef __attribute__((ext_vector_type(2))) float v2f;
typedef __attribute__((ext_vector_type(8))) float v8f;

// Fast reciprocal: single v_rcp_f32 instead of IEEE div sequence.
__device__ __forceinline__ float fast_rcp(float x) {
#if __has_builtin(__builtin_amdgcn_rcpf)
    return __builtin_amdgcn_rcpf(x);
#else
    return 1.0f / x;
#endif
}

// ---------------------------------------------------------------------------
// Kernel 0: zero the (B,K,A) accumulator in workspace
// ---------------------------------------------------------------------------
__global__ void zero_kernel(float* __restrict__ p, int n) {
    int i = blockIdx.x * blockDim.x + threadIdx.x;
    if (i < n) p[i] = 0.0f;
}

// ---------------------------------------------------------------------------
// Kernel 1: mask_area[b,a] = max(sum_hw masks[b,a,:,:], 1e-6)
// One block per (b,a) plane, 256-thread strided sum + LDS tree reduce.
// ---------------------------------------------------------------------------
__global__ __launch_bounds__(256) void mask_area_kernel(
    const float* __restrict__ masks, float* __restrict__ area) {
    const int pair = blockIdx.x;                    // b*AA + a
    const float* m = masks + (size_t)pair * HWSZ;
    float s = 0.0f;
    for (int i = threadIdx.x; i < HWSZ; i += 256) s += m[i];
    __shared__ float red[256];
    red[threadIdx.x] = s;
    __syncthreads();
    for (int off = 128; off > 0; off >>= 1) {
        if (threadIdx.x < off) red[threadIdx.x] += red[threadIdx.x + off];
        __syncthreads();
    }
    if (threadIdx.x == 0) area[pair] = fmaxf(red[0], 1e-6f);
}

// ---------------------------------------------------------------------------
// Kernel 2: fused decode (WMMA f32 16x16x4) + sigmoid + sq-err + masked MAC.
// grid = (tileGroups, K). 8 waves/block: wave id bit0 = batch half (mt),
// bits[2:1] = tile stream. Per 16-pixel tile:
//   D(16b x 16hw) = sum_p A(16b x p) * Wdec(p x 16hw) + b_dec   (4 WMMAs/chan)
//   err[b,hw]     = sum_c (sigmoid(D_c) - img[b,c,hw])^2
//   acc[a][r]    += err * mask[b,a,hw]        (per-lane, no shuffles in loop)
// End of wave: 16-lane half reduction via shfl_xor, atomicAdd into sums.
// ---------------------------------------------------------------------------
#if __has_builtin(__builtin_amdgcn_wmma_f32_16x16x4_f32)
#define HAVE_WMMA_F32 1
#else
#define HAVE_WMMA_F32 0
#endif

__global__ __launch_bounds__(256) void fused_cost_kernel(
    const float* __restrict__ poses,    // (B,K,P)
    const float* __restrict__ images,   // (B,C,H,W)
    const float* __restrict__ masks,    // (B,A,H,W)
    const float* __restrict__ Wdec,     // (K,P,O)
    const float* __restrict__ bdec,     // (O,)
    float* __restrict__ sums)           // (B,K,A) atomic accumulator
{
    const int k    = blockIdx.y;
    const int tid  = threadIdx.x;
    const int lane = tid & 31;
    const int wid  = tid >> 5;
    const int mt   = wid & 1;          // batch half: rows b = mt*16 .. mt*16+15
    const int strm = wid >> 1;         // tile stream 0..3
    const int lh   = lane >> 4;        // lane half (0: lanes 0-15, 1: 16-31)
    const int col  = lane & 15;        // tile column / row-in-16 for A loads

    // --- A-matrix tiles (poses) for this (mt,k): 4 K-chunks x 2 VGPRs ------
    // f32 A 16x4 layout: VGPR0 = {K=4j  (lanes0-15), K=4j+2 (lanes16-31)},
    //                    VGPR1 = {K=4j+1,            K=4j+3}; M = lane%16.
    float a0[4], a1[4];
    {
        const int b = mt * 16 + col;
        const float* pb = poses + ((size_t)b * KK + k) * PP;
        #pragma unroll
        for (int j = 0; j < 4; ++j) {
            const int p0 = 4 * j + (lh ? 2 : 0);
            a0[j] = pb[p0];
            a1[j] = pb[p0 + 1];
        }
    }

    float acc[AA][8];
    #pragma unroll
    for (int a = 0; a < AA; ++a)
        #pragma unroll
        for (int r = 0; r < 8; ++r) acc[a][r] = 0.0f;

    const int nstreams = gridDim.x * 4;
    for (int t = blockIdx.x * 4 + strm; t < NTILES; t += nstreams) {
        const int hw0 = t * 16;

        // --- B-matrix tiles (W_dec): per channel, 4 K-chunks x 2 VGPRs ----
        // f32 B 4x16 layout: VGPR0 = {row 4j (lanes0-15), row 4j+2 (16-31)},
        //                    VGPR1 = {row 4j+1, row 4j+3}; N = lane%16.
        v2f bw[CC][4];
        #pragma unroll
        for (int c = 0; c < CC; ++c) {
            #pragma unroll
            for (int j = 0; j < 4; ++j) {
                const int p0 = 4 * j + (lh ? 2 : 0);
                const float* wb =
                    Wdec + ((size_t)(k * PP + p0)) * OO + c * HWSZ + hw0 + col;
                bw[c][j].x = wb[0];
                bw[c][j].y = wb[OO];   // next p row
            }
        }
        float bd[CC];
        #pragma unroll
        for (int c = 0; c < CC; ++c) bd[c] = bdec[c * HWSZ + hw0 + col];

        // --- decode + sigmoid + squared error summed over channels --------
        float errsum[8];
        #pragma unroll
        for (int r = 0; r < 8; ++r) errsum[r] = 0.0f;

        #pragma unroll
        for (int c = 0; c < CC; ++c) {
            v8f d;
            #pragma unroll
            for (int r = 0; r < 8; ++r) d[r] = bd[c];   // C seeded with b_dec
#if HAVE_WMMA_F32
            #pragma unroll
            for (int j = 0; j < 4; ++j) {
                v2f av; av.x = a0[j]; av.y = a1[j];
                d = __builtin_amdgcn_wmma_f32_16x16x4_f32(
                        false, av, false, bw[c][j], (short)0, d, false, false);
            }
#else
            // Fallback (no WMMA builtin): per-lane FMA over broadcast A rows.
            #pragma unroll
            for (int j = 0; j < 4; ++j) {
                #pragma unroll
                for (int r = 0; r < 8; ++r) {
                    const int m  = r + lh * 8;
                    const int bb = mt * 16 + m;
                    const float* pb = poses + ((size_t)bb * KK + k) * PP + 4 * j;
                    const float* wb = Wdec + ((size_t)(k * PP + 4 * j)) * OO +
                                      c * HWSZ + hw0 + col;
                    #pragma unroll
                    for (int q = 0; q < 4; ++q)
                        d[r] = fmaf(pb[q], wb[(size_t)q * OO], d[r]);
                }
            }
#endif
            // C/D layout: VGPR r -> row M = r + 8*laneHalf, col N = lane%16
            const float* ib = images +
                ((size_t)(mt * 16 + lh * 8)) * (CC * HWSZ) + c * HWSZ + hw0 + col;
            #pragma unroll
            for (int r = 0; r < 8; ++r) {
                // sigmoid via v_exp_f32 + v_rcp_f32 (no IEEE div sequence)
                const float sg = fast_rcp(1.0f + __expf(-d[r]));
                const float df = sg - ib[(size_t)r * (CC * HWSZ)];
                errsum[r] = fmaf(df, df, errsum[r]);
            }
        }

        // --- masked MAC into per-lane accumulators ------------------------
        const float* mb = masks +
            ((size_t)(mt * 16 + lh * 8)) * (AA * HWSZ) + hw0 + col;
        #pragma unroll
        for (int a = 0; a < AA; ++a) {
            #pragma unroll
            for (int r = 0; r < 8; ++r) {
                const float mv = mb[(size_t)r * (AA * HWSZ) + (size_t)a * HWSZ];
                acc[a][r] = fmaf(errsum[r], mv, acc[a][r]);
            }
        }
    }

    // --- wave-end reduction over the 16 hw columns of each lane half ------
    #pragma unroll
    for (int a = 0; a < AA; ++a) {
        #pragma unroll
        for (int r = 0; r < 8; ++r) {
            float v = acc[a][r];
            v += __shfl_xor(v, 1);
            v += __shfl_xor(v, 2);
            v += __shfl_xor(v, 4);
            v += __shfl_xor(v, 8);
            if (col == 0) {
                const int b = mt * 16 + lh * 8 + r;
                atomicAdd(&sums[((size_t)b * KK + k) * AA + a], v);
            }
        }
    }
}

// ---------------------------------------------------------------------------
// Kernel 3: out = BCE-with-logits cost + masked recon cost
// ---------------------------------------------------------------------------
__global__ void finalize_kernel(
    const float* __restrict__ logits,   // (B,K)
    const int*   __restrict__ targets,  // (B,A)
    const float* __restrict__ sums,     // (B,K,A)
    const float* __restrict__ area,     // (B,A)
    float* __restrict__ out)            // (B,K,A)
{
    const int i = blockIdx.x * blockDim.x + threadIdx.x;
    if (i >= BB * KK * AA) return;
    const int a = i % AA;
    const int k = (i / AA) % KK;
    const int b = i / (KK * AA);
    const float l = logits[b * KK + k];
    const float t = (float)targets[b * AA + a];
    // logaddexp(0, l) = max(l,0) + log1p(exp(-|l|))
    const float cls = fmaxf(l, 0.0f) + log1pf(__expf(-fabsf(l))) - l * t;
    float rec = sums[i] / area[b * AA + a];
    rec = (t > 0.5f) ? rec : 0.0f;
    out[i] = cls + rec;
}

// ---------------------------------------------------------------------------
// Host launch
// ---------------------------------------------------------------------------
extern "C" void kernel_launch(void* const* d_in, const int* in_sizes, int n_in,
                              void* d_out, int out_size, void* d_ws, size_t ws_size,
                              hipStream_t stream) {
    const float* logits  = (const float*)d_in[0];  // (B,K)
    const float* poses   = (const float*)d_in[1];  // (B,K,P)
    const float* images  = (const float*)d_in[2];  // (B,C,H,W)
    const float* masks   = (const float*)d_in[3];  // (B,A,H,W)
    const float* Wdec    = (const float*)d_in[4];  // (K,P,O)
    const float* bdec    = (const float*)d_in[5];  // (O,)
    const int*   targets = (const int*)d_in[6];    // (B,A)
    float* out = (float*)d_out;

    float* sums = (float*)d_ws;          // B*K*A = 2048 floats
    float* area = sums + BB * KK * AA;   // B*A   =  256 floats

    zero_kernel<<<(BB * KK * AA + 255) / 256, 256, 0, stream>>>(sums, BB * KK * AA);
    mask_area_kernel<<<BB * AA, 256, 0, stream>>>(masks, area);

    dim3 grid(50, KK);                   // 50*4 tile streams x 8 capsules
    fused_cost_kernel<<<grid, 256, 0, stream>>>(poses, images, masks, Wdec,
                                                bdec, sums);

    finalize_kernel<<<(BB * KK * AA + 255) / 256, 256, 0, stream>>>(
        logits, targets, sums, area, out);
}